// MultiLayerGATFeatureAggregator_7267084665516
// MI455X (gfx1250) — compile-verified
//
#include <hip/hip_runtime.h>
#include <hip/hip_bf16.h>
#include <stdint.h>

typedef __attribute__((ext_vector_type(2))) float v2f;
typedef __attribute__((ext_vector_type(8))) float v8f;

#define N_NODES 50000
#define N_EDGES 800000
#define E_TOT   (N_EDGES + N_NODES)
#define IN_DIM  128
#define HID     64
#define HEADS   4
#define C1      (HEADS * HID)   /* 256 */
#define OUT_DIM 128

__device__ __forceinline__ float lrelu(float v) {
  return fmaxf(v, 0.f) + 0.2f * fminf(v, 0.f);
}

// float atomic max via sign-aware integer atomics (monotone mapping trick)
__device__ __forceinline__ void atomicMaxF(float* addr, float val) {
  if (val >= 0.f) atomicMax((int*)addr, __float_as_int(val));
  else            atomicMin((unsigned int*)addr, __float_as_uint(val));
}

__global__ void k_fill(float* __restrict__ p, float v, int n) {
  int i = blockIdx.x * blockDim.x + threadIdx.x;
  if (i < n) p[i] = v;
}

// C[M,Kout] = A[M,K] @ W[K,Kout], f32, one wave per 16x16 tile.
// M is a multiple of 16 (50000 = 3125*16) so no masking: EXEC all-ones for WMMA.
// A-frag (16x4): lanes 0-15 hold K=k0+{0,1}, lanes 16-31 hold K=k0+{2,3} (ISA 7.12.2)
// B-frag (4x16): symmetric; C/D: row = vgpr + 8*(lane>=16), col = lane&15
__global__ void k_gemm_wmma(const float* __restrict__ A, const float* __restrict__ W,
                            float* __restrict__ C, int K, int Kout, int n_tiles) {
  int wid = (blockIdx.x * blockDim.x + threadIdx.x) >> 5;
  if (wid >= n_tiles) return;                 // wave-uniform exit
  int ntiles = Kout >> 4;
  int mt = wid / ntiles;
  int nt = wid - mt * ntiles;
  int lane = threadIdx.x & 31;
  int hh = lane >> 4;                         // half-wave selector
  int lr = lane & 15;
  const float* arow = A + (size_t)(mt * 16 + lr) * K + 2 * hh;
  const float* bcol = W + (size_t)(2 * hh) * Kout + nt * 16 + lr;
  v8f acc = {};
  for (int k0 = 0; k0 < K; k0 += 4) {
    v2f a, b;
    a.x = arow[k0];
    a.y = arow[k0 + 1];
    b.x = bcol[(size_t)k0 * Kout];
    b.y = bcol[(size_t)(k0 + 1) * Kout];
    acc = __builtin_amdgcn_wmma_f32_16x16x4_f32(false, a, false, b,
                                                (short)0, acc, false, false);
  }
  float* crow = C + (size_t)(mt * 16 + 8 * hh) * Kout + nt * 16 + lr;
#pragma unroll
  for (int v = 0; v < 8; ++v) crow[(size_t)v * Kout] = acc[v];
}

// per-(node,head) attention logits: as[i] = <h[i,:], a_src[h,:]>, ad likewise
__global__ void k_alpha(const float* __restrict__ Hm, const float* __restrict__ a_src,
                        const float* __restrict__ a_dst, float* __restrict__ as_,
                        float* __restrict__ ad_, int n_total, int H, int Ch) {
  int i = blockIdx.x * blockDim.x + threadIdx.x;
  if (i >= n_total) return;
  int h = i % H;
  const float4* hr = (const float4*)(Hm + (size_t)i * Ch);
  const float4* sr = (const float4*)(a_src + (size_t)h * Ch);
  const float4* dr = (const float4*)(a_dst + (size_t)h * Ch);
  float s = 0.f, d = 0.f;
  for (int c = 0; c < (Ch >> 2); ++c) {
    float4 hv = hr[c], sv = sr[c], dv = dr[c];
    s += hv.x * sv.x + hv.y * sv.y + hv.z * sv.z + hv.w * sv.w;
    d += hv.x * dv.x + hv.y * dv.y + hv.z * dv.z + hv.w * dv.w;
  }
  as_[i] = s;
  ad_[i] = d;
}

// segment max of leaky_relu(as[src]+ad[dst]) into mx[dst,h]
__global__ void k_edge_max(const int* __restrict__ ei, const float* __restrict__ as_,
                           const float* __restrict__ ad_, float* __restrict__ mx,
                           int hShift, int total) {
  int i = blockIdx.x * blockDim.x + threadIdx.x;
  if (i >= total) return;
  int e = i >> hShift;
  int h = i & ((1 << hShift) - 1);
  int H = 1 << hShift;
  int s, d;
  if (e < N_EDGES) { s = ei[e]; d = ei[N_EDGES + e]; } else { s = d = e - N_EDGES; }
  float v = lrelu(as_[s * H + h] + ad_[d * H + h]);
  atomicMaxF(&mx[d * H + h], v);
}

// ex = exp(e - mx[dst]); store per-edge, accumulate softmax denominator
__global__ void k_edge_exp(const int* __restrict__ ei, const float* __restrict__ as_,
                           const float* __restrict__ ad_, const float* __restrict__ mx,
                           float* __restrict__ exb, float* __restrict__ dn,
                           int hShift, int total) {
  int i = blockIdx.x * blockDim.x + threadIdx.x;
  if (i >= total) return;
  int e = i >> hShift;
  int h = i & ((1 << hShift) - 1);
  int H = 1 << hShift;
  int s, d;
  if (e < N_EDGES) { s = ei[e]; d = ei[N_EDGES + e]; } else { s = d = e - N_EDGES; }
  float v = lrelu(as_[s * H + h] + ad_[d * H + h]);
  float ex = __expf(v - mx[d * H + h]);
  exb[i] = ex;
  atomicAdd(&dn[d * H + h], ex);
}

// out[dst,c] += (ex/denom) * h[src,c]; one wave per edge, 32-lane channel sweep
__global__ void k_edge_aggr(const int* __restrict__ ei, const float* __restrict__ Hm,
                            const float* __restrict__ exb, const float* __restrict__ dn,
                            float* __restrict__ Out, int hShift, int chShift, int Ct) {
  int wid = (blockIdx.x * blockDim.x + threadIdx.x) >> 5;
  if (wid >= E_TOT) return;                   // wave-uniform exit
  int lane = threadIdx.x & 31;
  int H = 1 << hShift;
  int s, d;
  if (wid < N_EDGES) { s = ei[wid]; d = ei[N_EDGES + wid]; } else { s = d = wid - N_EDGES; }
  const float* hs = Hm + (size_t)s * Ct;
  float* od = Out + (size_t)d * Ct;
  for (int c = lane; c < Ct; c += 32) {
    int h = c >> chShift;
    float w = exb[(size_t)wid * H + h] / dn[d * H + h];
    atomicAdd(&od[c], w * hs[c]);
  }
}

// in-place-safe: out[i] = leaky_relu(in[i] + bias[i & cmask])
__global__ void k_act(const float* __restrict__ in, const float* __restrict__ bias,
                      float* __restrict__ out, int cmask, int total) {
  int i = blockIdx.x * blockDim.x + threadIdx.x;
  if (i >= total) return;
  out[i] = lrelu(in[i] + bias[i & cmask]);
}

// out[i] = leaky_relu(in[i] + bias) + feat[i]  (identity residual)
__global__ void k_final(const float* __restrict__ in, const float* __restrict__ bias,
                        const float* __restrict__ feat, float* __restrict__ out, int total) {
  int i = blockIdx.x * blockDim.x + threadIdx.x;
  if (i >= total) return;
  out[i] = lrelu(in[i] + bias[i & (OUT_DIM - 1)]) + feat[i];
}

extern "C" void kernel_launch(void* const* d_in, const int* in_sizes, int n_in,
                              void* d_out, int out_size, void* d_ws, size_t ws_size,
                              hipStream_t stream) {
  const float* feat = (const float*)d_in[0];
  const int*   ei   = (const int*)d_in[1];
  const float* W1   = (const float*)d_in[2];
  const float* aS1  = (const float*)d_in[3];
  const float* aD1  = (const float*)d_in[4];
  const float* b1   = (const float*)d_in[5];
  const float* W2   = (const float*)d_in[6];
  const float* aS2  = (const float*)d_in[7];
  const float* aD2  = (const float*)d_in[8];
  const float* b2   = (const float*)d_in[9];
  float* out = (float*)d_out;

  // carve workspace (~172 MB of f32)
  float* p    = (float*)d_ws;
  float* h1   = p; p += (size_t)N_NODES * C1;       // layer1 features
  float* xo1  = p; p += (size_t)N_NODES * C1;       // layer1 accum -> activated x (in place)
  float* h2   = p; p += (size_t)N_NODES * OUT_DIM;  // layer2 features
  float* out2 = p; p += (size_t)N_NODES * OUT_DIM;  // layer2 accum
  float* as1  = p; p += (size_t)N_NODES * HEADS;
  float* ad1  = p; p += (size_t)N_NODES * HEADS;
  float* m1   = p; p += (size_t)N_NODES * HEADS;
  float* dn1  = p; p += (size_t)N_NODES * HEADS;
  float* as2  = p; p += (size_t)N_NODES;
  float* ad2  = p; p += (size_t)N_NODES;
  float* m2   = p; p += (size_t)N_NODES;
  float* dn2  = p; p += (size_t)N_NODES;
  float* ex1  = p; p += (size_t)E_TOT * HEADS;
  float* ex2  = p; p += (size_t)E_TOT;

  const int T = 256;
  const float NEG = -3.402823466e38f;
#define GRID(n) (((n) + T - 1) / T)

  // ---------------- layer 1 ----------------
  int tiles1 = (N_NODES / 16) * (C1 / 16);          // 50000 wave-tiles
  k_gemm_wmma<<<GRID(tiles1 * 32), T, 0, stream>>>(feat, W1, h1, IN_DIM, C1, tiles1);
  k_alpha<<<GRID(N_NODES * HEADS), T, 0, stream>>>(h1, aS1, aD1, as1, ad1,
                                                   N_NODES * HEADS, HEADS, HID);
  k_fill<<<GRID(N_NODES * C1), T, 0, stream>>>(xo1, 0.f, N_NODES * C1);
  k_fill<<<GRID(N_NODES * HEADS), T, 0, stream>>>(m1, NEG, N_NODES * HEADS);
  k_fill<<<GRID(N_NODES * HEADS), T, 0, stream>>>(dn1, 0.f, N_NODES * HEADS);
  k_edge_max<<<GRID(E_TOT * HEADS), T, 0, stream>>>(ei, as1, ad1, m1, 2, E_TOT * HEADS);
  k_edge_exp<<<GRID(E_TOT * HEADS), T, 0, stream>>>(ei, as1, ad1, m1, ex1, dn1, 2, E_TOT * HEADS);
  k_edge_aggr<<<GRID(E_TOT * 32), T, 0, stream>>>(ei, h1, ex1, dn1, xo1, 2, 6, C1);
  k_act<<<GRID(N_NODES * C1), T, 0, stream>>>(xo1, b1, xo1, C1 - 1, N_NODES * C1);

  // ---------------- layer 2 ----------------
  int tiles2 = (N_NODES / 16) * (OUT_DIM / 16);     // 25000 wave-tiles
  k_gemm_wmma<<<GRID(tiles2 * 32), T, 0, stream>>>(xo1, W2, h2, C1, OUT_DIM, tiles2);
  k_alpha<<<GRID(N_NODES), T, 0, stream>>>(h2, aS2, aD2, as2, ad2, N_NODES, 1, OUT_DIM);
  k_fill<<<GRID(N_NODES * OUT_DIM), T, 0, stream>>>(out2, 0.f, N_NODES * OUT_DIM);
  k_fill<<<GRID(N_NODES), T, 0, stream>>>(m2, NEG, N_NODES);
  k_fill<<<GRID(N_NODES), T, 0, stream>>>(dn2, 0.f, N_NODES);
  k_edge_max<<<GRID(E_TOT), T, 0, stream>>>(ei, as2, ad2, m2, 0, E_TOT);
  k_edge_exp<<<GRID(E_TOT), T, 0, stream>>>(ei, as2, ad2, m2, ex2, dn2, 0, E_TOT);
  k_edge_aggr<<<GRID(E_TOT * 32), T, 0, stream>>>(ei, h2, ex2, dn2, out2, 0, 7, OUT_DIM);
  k_final<<<GRID(N_NODES * OUT_DIM), T, 0, stream>>>(out2, b2, feat, out, N_NODES * OUT_DIM);
#undef GRID
}